// JointAngleLoss_51677046505851
// MI455X (gfx1250) — compile-verified
//
#include <hip/hip_runtime.h>
#include <hip/hip_bf16.h>

#ifndef __has_builtin
#define __has_builtin(x) 0
#endif

// ---------- CDNA5 feature probes (warnings show up in compile stderr) ----------
#if __has_builtin(__builtin_amdgcn_global_load_async_to_lds_b128)
#define HAVE_ASYNC_LDS 1
#else
#define HAVE_ASYNC_LDS 0
#warning "CDNA5 probe: __builtin_amdgcn_global_load_async_to_lds_b128 NOT available -> plain staging loads"
#endif

#if __has_builtin(__builtin_amdgcn_s_wait_asynccnt)
#define WAIT_ASYNC() __builtin_amdgcn_s_wait_asynccnt(0)
#else
#define WAIT_ASYNC() asm volatile("s_wait_asynccnt 0" ::: "memory")
#endif

#if __has_builtin(__builtin_amdgcn_wmma_f32_16x16x4_f32)
#define HAVE_WMMA 1
#else
#define HAVE_WMMA 0
#warning "CDNA5 probe: __builtin_amdgcn_wmma_f32_16x16x4_f32 NOT available -> scalar finish"
#endif

typedef float v2f __attribute__((ext_vector_type(2)));
typedef float v8f __attribute__((ext_vector_type(8)));
typedef int   v4i __attribute__((ext_vector_type(4)));

// Address-space-qualified pointer types for the async-LDS builtin
// (clang's diagnostic showed param 0 as AS1 `int4*`; AS3 for the LDS side).
typedef __attribute__((address_space(1))) v4i* as1_v4i_ptr;
typedef __attribute__((address_space(3))) v4i* as3_v4i_ptr;

// Per-finger loss: p points at joint 4f (15 consecutive floats = joints 4f..4f+4)
__device__ __forceinline__ float finger_loss(const float* __restrict__ p) {
    float b1x = p[3]  - p[0],  b1y = p[4]  - p[1],  b1z = p[5]  - p[2];
    float b2x = p[6]  - p[3],  b2y = p[7]  - p[4],  b2z = p[8]  - p[5];
    float b3x = p[9]  - p[6],  b3y = p[10] - p[7],  b3z = p[11] - p[8];
    float b4x = p[12] - p[9],  b4y = p[13] - p[10], b4z = p[14] - p[11];
    // r_tip = b4 x b3
    float rtx = b4y*b3z - b4z*b3y, rty = b4z*b3x - b4x*b3z, rtz = b4x*b3y - b4y*b3x;
    // r_mid = b3 x b2
    float rmx = b3y*b2z - b3z*b2y, rmy = b3z*b2x - b3x*b2z, rmz = b3x*b2y - b3y*b2x;
    // r_palm = b2 x b1
    float rpx = b2y*b1z - b2z*b1y, rpy = b2z*b1x - b2x*b1z, rpz = b2x*b1y - b2y*b1x;
    float mn = rpx*b4x + rpy*b4y + rpz*b4z + rmx*b4x + rmy*b4y + rmz*b4z;
    float d1 = rtx*rmx + rty*rmy + rtz*rmz;
    float d2 = rpx*rmx + rpy*rmy + rpz*rmz;
    float pen = (d1 < 0.f ? d1*d1 : 0.f) + (d2 < 0.f ? d2*d2 : 0.f);
    return mn + pen;
}

// 256 threads/block = 8 waves; 256 hands/block; 63 KB LDS staging.
// Chunk per block = 256*63*4 = 64512 bytes = 4032 float4 (16B aligned: 64512 % 16 == 0).
__global__ __launch_bounds__(256) void jal_main(const float* __restrict__ gt,
                                                float* __restrict__ partial) {
    __shared__ __align__(16) float lds[16128];
    const int tid = threadIdx.x;
    const int bid = blockIdx.x;
    const float4* __restrict__ src = (const float4*)gt + (size_t)bid * 4032;

#if HAVE_ASYNC_LDS
    for (int i = tid; i < 4032; i += 256) {
        float4* s = const_cast<float4*>(src + i);
        float*  d = lds + 4 * i;
        __builtin_amdgcn_global_load_async_to_lds_b128(
            (as1_v4i_ptr)s, (as3_v4i_ptr)d, /*offset=*/0, /*cpol=*/0);
    }
    WAIT_ASYNC();
    __syncthreads();
#else
    {
        float4* ldsv = (float4*)lds;
        for (int i = tid; i < 4032; i += 256) ldsv[i] = src[i];
        __syncthreads();
    }
#endif

    // Each thread computes loss of one hand from LDS (stride 63 -> bank-conflict-free).
    const float* __restrict__ h = &lds[tid * 63];
    float loss = 0.f;
    #pragma unroll
    for (int f = 0; f < 5; ++f) loss += finger_loss(h + 12 * f);

    // Deterministic block tree reduction (reuse staging LDS after a barrier).
    __syncthreads();
    lds[tid] = loss;
    __syncthreads();
    #pragma unroll
    for (int s = 128; s > 0; s >>= 1) {
        if (tid < s) lds[tid] += lds[tid + s];
        __syncthreads();
    }
    if (tid == 0) partial[bid] = lds[0];
}

// Single wave32 finish. WMMA path: A = ones(16x4), B carries 64 partials per issue,
// C accumulates; every row of D = column sums, so lane L's c.x (L<16) holds a column
// sum; 16-lane shfl tree yields the total. Fixed-order => deterministic.
__global__ __launch_bounds__(32) void jal_reduce(const float* __restrict__ partial,
                                                 float* __restrict__ out, int n) {
#if HAVE_WMMA
    const int lane = threadIdx.x;
    v2f a; a.x = 1.f; a.y = 1.f;
    v8f c = {};
    for (int base = 0; base < n; base += 64) {
        v2f b;
        b.x = partial[base + lane];
        b.y = partial[base + 32 + lane];
        c = __builtin_amdgcn_wmma_f32_16x16x4_f32(
                /*neg_a=*/false, a, /*neg_b=*/false, b,
                /*c_mod=*/(short)0, c, /*reuse_a=*/false, /*reuse_b=*/false);
    }
    float v = c.x;                      // lanes 0..15: D[0][c]; 16..31: identical row 8
    v += __shfl_xor(v, 8, 32);
    v += __shfl_xor(v, 4, 32);
    v += __shfl_xor(v, 2, 32);
    v += __shfl_xor(v, 1, 32);          // lane 0 = sum over 16 columns = grand total
    if (lane == 0) *out = v;
#else
    if (threadIdx.x == 0) {
        float s = 0.f;
        for (int i = 0; i < n; ++i) s += partial[i];
        *out = s;
    }
#endif
}

extern "C" void kernel_launch(void* const* d_in, const int* in_sizes, int n_in,
                              void* d_out, int out_size, void* d_ws, size_t ws_size,
                              hipStream_t stream) {
    (void)n_in; (void)out_size; (void)ws_size;
    // d_in[0] = pose23d_pred (UNUSED by the reference), d_in[1] = pose23d_gt
    const float* gt = (const float*)d_in[1];
    const int B = in_sizes[1] / 63;           // 524288 hands
    const int blocks = B / 256;               // 2048 (B is a multiple of 256)
    float* partial = (float*)d_ws;            // 2048 floats scratch

    jal_main<<<blocks, 256, 0, stream>>>(gt, partial);
    jal_reduce<<<1, 32, 0, stream>>>(partial, (float*)d_out, blocks);
}